// GammaSpaceLayer_50208167690595
// MI455X (gfx1250) — compile-verified
//
#include <hip/hip_runtime.h>

typedef float v2f __attribute__((ext_vector_type(2)));
typedef float v8f __attribute__((ext_vector_type(8)));
typedef unsigned int u32x4 __attribute__((ext_vector_type(4)));
typedef int i32x8 __attribute__((ext_vector_type(8)));
typedef int i32x4 __attribute__((ext_vector_type(4)));

#define HDIM   64      // HIDDEN_DIM (state size)
#define SDIM   128     // STATE_DIM (channels)
#define NBATCH 16
#define SEQL   2048
#define TCHK   64      // chunk length
#define NCHK   32      // SEQL / TCHK
#define NCOL   (NBATCH * NCHK)   // 512 independent scan columns
#define NTOT   (NCOL * TCHK)     // 32768 total time-columns

// ---- workspace layout (floats) ----
#define WS_DA    0                         // 64*64      = 4096
#define WS_DB    4096                      // 64*128     = 8192
#define WS_P     12288                     // 64*64      = 4096
#define WS_W     16384                     // 64*128*64  = 524288  (W[t] = C*dA^(t+1), (t,o,h))
#define WS_SG    540672                    // NTOT*64    = 2097152 (S_intra, layout (gc,t,h))
#define WS_SIN   2637824                   // 64*NCOL    = 32768   (SinT[h][b*32+c])
#define WS_V     2670592                   // NTOT*64    = 2097152 (V = dB*u, layout (gc,t,h))

// fp32 WMMA 16x16x4: D(16x16) += A(16x4) * B(4x16)
__device__ __forceinline__ v8f wmma_f32_4(v2f a, v2f b, v8f c) {
  return __builtin_amdgcn_wmma_f32_16x16x4_f32(false, a, false, b, (short)0, c, false, false);
}

// ---------------------------------------------------------------------------
// K1: discretize.  backward = (1+dt/2)I - (dt/2)N,  forward = (1-dt/2)I + (dt/2)N
// dA = backward^-1 forward, dB = backward^-1 (dt*B)  (forward substitution)
// ---------------------------------------------------------------------------
__global__ void gsl_setup(const float* __restrict__ Bmat,
                          const float* __restrict__ logdt,
                          float* __restrict__ dA, float* __restrict__ dB) {
  const int tid = threadIdx.x;
  float x = logdt[0];
  float dt = (x > 20.f) ? x : log1pf(expf(x));           // softplus
  dt = fminf(fmaxf(dt, 0.001f), 0.1f);
  const float b1 = -0.5f * dt;
  const float f0 = 1.f - 0.5f * dt, f1 = 0.5f * dt;
  const float inv = 1.f / (1.f + 0.5f * dt);
  if (tid < 64) {                // dA column j
    const int j = tid;
    float xp = 0.f;
    for (int i = 0; i < 64; ++i) {
      float f = (i == j) ? f0 : ((i == j + 1) ? f1 : 0.f);
      float xv = (f - b1 * xp) * inv;
      dA[i * 64 + j] = xv;
      xp = xv;
    }
  }
  if (tid < 128) {               // dB column j
    const int j = tid;
    float xp = 0.f;
    for (int i = 0; i < 64; ++i) {
      float rhs = dt * Bmat[i * 128 + j];
      float xv = (rhs - b1 * xp) * inv;
      dB[i * 128 + j] = xv;
      xp = xv;
    }
  }
}

// ---------------------------------------------------------------------------
// K2: block t computes dA^(t+1) by binary powering, then W[t] = C * dA^(t+1).
// Block 63 also stores P = dA^64.
// ---------------------------------------------------------------------------
__device__ void mm64(const float* A, const float* B, float* D, int tid) {
  for (int idx = tid; idx < 64 * 64; idx += 128) {
    int i = idx >> 6, j = idx & 63;
    float s = 0.f;
    for (int m = 0; m < 64; ++m) s += A[i * 64 + m] * B[m * 64 + j];
    D[idx] = s;
  }
}

__global__ void gsl_powers(const float* __restrict__ dA,
                           const float* __restrict__ Cmat,
                           float* __restrict__ Wst, float* __restrict__ P) {
  __shared__ float base[4096], res[4096], tmp[4096];
  const int tid = threadIdx.x;
  for (int idx = tid; idx < 4096; idx += 128) {
    base[idx] = dA[idx];
    res[idx] = ((idx >> 6) == (idx & 63)) ? 1.f : 0.f;
  }
  __syncthreads();
  int e = (int)blockIdx.x + 1;   // power 1..64
  while (e) {
    if (e & 1) {
      mm64(res, base, tmp, tid); __syncthreads();
      for (int idx = tid; idx < 4096; idx += 128) res[idx] = tmp[idx];
      __syncthreads();
    }
    e >>= 1;
    if (e) {
      mm64(base, base, tmp, tid); __syncthreads();
      for (int idx = tid; idx < 4096; idx += 128) base[idx] = tmp[idx];
      __syncthreads();
    }
  }
  float* wt = Wst + (size_t)blockIdx.x * (128 * 64);
  for (int idx = tid; idx < 128 * 64; idx += 128) {
    int o = idx >> 6, h = idx & 63;
    float s = 0.f;
    for (int m = 0; m < 64; ++m) s += Cmat[o * 64 + m] * res[m * 64 + h];
    wt[idx] = s;
  }
  if (blockIdx.x == 63)
    for (int idx = tid; idx < 4096; idx += 128) P[idx] = res[idx];
}

// ---------------------------------------------------------------------------
// K3: V = dB (64x128) * u  — fully parallel streaming WMMA (no serial dep).
// V layout: V[(b*SEQL + l)*64 + h]  == (gc, t, h)
// ---------------------------------------------------------------------------
__global__ void gsl_v(const float* __restrict__ dB,
                      const float* __restrict__ u,
                      float* __restrict__ V) {
  const int tid = threadIdx.x;
  const int wave = tid >> 5, lane = tid & 31;
  const int row = lane & 15, hi = lane >> 4;
  const int m0 = wave * 16;
  const int b = blockIdx.y;
  const int l0 = blockIdx.x * 16;
  const float* urow = u + (size_t)(b * SEQL + l0 + row) * SDIM;
  v8f acc = {};
#pragma unroll
  for (int k0 = 0; k0 < 128; k0 += 4) {
    int kk = k0 + 2 * hi;
    v2f a; a.x = dB[(m0 + row) * 128 + kk]; a.y = dB[(m0 + row) * 128 + kk + 1];
    v2f bb; bb.x = urow[kk]; bb.y = urow[kk + 1];
    acc = wmma_f32_4(a, bb, acc);
  }
  float* vb = V + (size_t)(b * SEQL + l0 + row) * 64 + m0 + 8 * hi;
  *(float4*)vb       = make_float4(acc[0], acc[1], acc[2], acc[3]);
  *(float4*)(vb + 4) = make_float4(acc[4], acc[5], acc[6], acc[7]);
}

// ---------------------------------------------------------------------------
// K4: intra-chunk scan.  64 sequential steps of  S <- dA*S + V_t  (WMMA, LDS).
// dA staged into LDS by the Tensor Data Mover (one D#, 16 KB tile).
// ---------------------------------------------------------------------------
__global__ void gsl_scan(const float* __restrict__ dA,
                         const float* __restrict__ V,
                         float* __restrict__ Sg) {
  __shared__ float dAl[64 * 64];
  __shared__ float Scur[64 * 16];
  const int tid = threadIdx.x;

  if (tid < 32) {  // wave 0 issues the TDM DMA: 4096 fp32, 1 row
    unsigned lds_off = (unsigned)(size_t)&dAl[0];
    unsigned long long ga = (unsigned long long)(size_t)dA;
    u32x4 g0 = { 1u,                                   // count=1, user mode
                 lds_off,                              // lds_addr
                 (unsigned)ga,                         // global_addr[31:0]
                 (unsigned)((ga >> 32) & 0x1FFFFFFull) | (2u << 30) }; // addr[56:32] | type=2
    i32x8 g1 = { 0x20000,       // data_size=2 (4B), no multicast
                 0x10000000,    // tensor_dim0 = 4096 (bits 63:48)
                 0x00010000,    // tensor_dim0 hi = 0, tensor_dim1 = 1
                 0x10000000,    // tile_dim0 = 4096 (bits 127:112)
                 1,             // tile_dim1 = 1
                 4096,          // tensor_dim0_stride = 4096
                 0, 0 };
    i32x4 z4 = { 0, 0, 0, 0 };
    i32x8 z8 = { 0, 0, 0, 0, 0, 0, 0, 0 };
    __builtin_amdgcn_tensor_load_to_lds(g0, g1, z4, z4, z8, 0);
    __builtin_amdgcn_s_wait_tensorcnt(0);
  }
  for (int idx = tid; idx < 1024; idx += 128) Scur[idx] = 0.f;
  __syncthreads();

  const int wave = tid >> 5, lane = tid & 31;
  const int row = lane & 15, hi = lane >> 4;
  const int m0 = wave * 16;
  const int gc = (int)blockIdx.x * 16 + row;      // this lane's (b,c) column
  const float4* vp = (const float4*)(V + (size_t)gc * TCHK * 64 + m0 + 8 * hi);

  for (int t = 0; t < TCHK; ++t) {
    v8f acc = {};
#pragma unroll
    for (int k0 = 0; k0 < 64; k0 += 4) {
      int kk = k0 + 2 * hi;
      v2f a; a.x = dAl[(m0 + row) * 64 + kk]; a.y = dAl[(m0 + row) * 64 + kk + 1];
      v2f b; b.x = Scur[kk * 16 + row];       b.y = Scur[(kk + 1) * 16 + row];
      acc = wmma_f32_4(a, b, acc);
    }
    float4 va = vp[t * 16], vb = vp[t * 16 + 1];
    acc[0] += va.x; acc[1] += va.y; acc[2] += va.z; acc[3] += va.w;
    acc[4] += vb.x; acc[5] += vb.y; acc[6] += vb.z; acc[7] += vb.w;
    __syncthreads();
#pragma unroll
    for (int r = 0; r < 8; ++r)
      Scur[(m0 + r + 8 * hi) * 16 + row] = acc[r];
    float* sb = Sg + ((size_t)gc * TCHK + t) * 64 + m0 + 8 * hi;
    *(float4*)sb       = make_float4(acc[0], acc[1], acc[2], acc[3]);
    *(float4*)(sb + 4) = make_float4(acc[4], acc[5], acc[6], acc[7]);
    __syncthreads();
  }
}

// ---------------------------------------------------------------------------
// K5: chunk-boundary scan.  s_in[b,c] = P * s_in[b,c-1] + S_intra[b,c-1,63].
// Stored transposed: SinT[h*NCOL + b*NCHK + c].
// ---------------------------------------------------------------------------
__global__ void gsl_chunks(const float* __restrict__ P,
                           const float* __restrict__ Sg,
                           float* __restrict__ SinT) {
  __shared__ float sprev[NBATCH * 64], snew[NBATCH * 64];
  const int tid = threadIdx.x;
  for (int idx = tid; idx < NBATCH * 64; idx += 256) {
    sprev[idx] = 0.f;
    int b = idx >> 6, h = idx & 63;
    SinT[h * NCOL + b * NCHK + 0] = 0.f;
  }
  __syncthreads();
  for (int c = 1; c < NCHK; ++c) {
    for (int idx = tid; idx < NBATCH * 64; idx += 256) {
      int b = idx >> 6, h = idx & 63;
      float s = Sg[((size_t)(b * NCHK + (c - 1)) * TCHK + (TCHK - 1)) * 64 + h];
      for (int m = 0; m < 64; ++m) s += P[h * 64 + m] * sprev[b * 64 + m];
      snew[idx] = s;
      SinT[h * NCOL + b * NCHK + c] = s;
    }
    __syncthreads();
    for (int idx = tid; idx < NBATCH * 64; idx += 256) sprev[idx] = snew[idx];
    __syncthreads();
  }
}

// ---------------------------------------------------------------------------
// K6: y = C*S_intra (WMMA) + W[t]*s_in (carry) + D.*u, streamed to d_out.
// grid: x = chunk*4 + t-quarter (128), y = o-half (2), z = batch (16)
// ---------------------------------------------------------------------------
__global__ void gsl_final(const float* __restrict__ u,
                          const float* __restrict__ Cmat,
                          const float* __restrict__ Dvec,
                          const float* __restrict__ Sg,
                          const float* __restrict__ SinT,
                          const float* __restrict__ Wst,
                          float* __restrict__ out) {
  const int cidx = blockIdx.x >> 2, tq = blockIdx.x & 3;
  const int b = blockIdx.z;
  __shared__ float sin_l[64];
  if (threadIdx.x < 64)
    sin_l[threadIdx.x] = SinT[threadIdx.x * NCOL + b * NCHK + cidx];
  __syncthreads();

  const int wave = threadIdx.x >> 5, lane = threadIdx.x & 31;
  const int row = lane & 15, hi = lane >> 4;
  const int o0 = blockIdx.y * 64 + wave * 16;
  const size_t colbase = ((size_t)(b * NCHK + cidx) * TCHK + tq * 16 + row) * 64;

  v8f acc = {};
#pragma unroll
  for (int k0 = 0; k0 < 64; k0 += 4) {
    int kk = k0 + 2 * hi;
    v2f a; a.x = Cmat[(o0 + row) * 64 + kk]; a.y = Cmat[(o0 + row) * 64 + kk + 1];
    v2f bb; bb.x = Sg[colbase + kk]; bb.y = Sg[colbase + kk + 1];
    acc = wmma_f32_4(a, bb, acc);
  }

  const int t_local = tq * 16 + row;          // this lane's time within chunk
  const int l = cidx * TCHK + t_local;
#pragma unroll
  for (int r = 0; r < 8; ++r) {
    const int o = o0 + r + 8 * hi;
    float v = acc[r];
    const float* wrow = Wst + ((size_t)(t_local * 128 + o) << 6);
    float cs = 0.f;
    for (int h = 0; h < 64; ++h) cs += wrow[h] * sin_l[h];
    v += cs;
    const float uval = u[((size_t)(b * SEQL + l) << 7) + o];
    v += Dvec[o] * uval;
    out[((size_t)(b * SEQL + l) << 7) + o] = v;
  }
}

// ---------------------------------------------------------------------------
extern "C" void kernel_launch(void* const* d_in, const int* in_sizes, int n_in,
                              void* d_out, int out_size, void* d_ws, size_t ws_size,
                              hipStream_t stream) {
  (void)in_sizes; (void)n_in; (void)out_size; (void)ws_size;
  const float* u     = (const float*)d_in[0];
  const float* Bmat  = (const float*)d_in[1];
  const float* Cmat  = (const float*)d_in[2];
  const float* Dvec  = (const float*)d_in[3];
  const float* logdt = (const float*)d_in[4];

  float* ws   = (float*)d_ws;
  float* dA   = ws + WS_DA;
  float* dB   = ws + WS_DB;
  float* P    = ws + WS_P;
  float* Wst  = ws + WS_W;
  float* Sg   = ws + WS_SG;
  float* SinT = ws + WS_SIN;
  float* V    = ws + WS_V;

  gsl_setup <<<1, 128, 0, stream>>>(Bmat, logdt, dA, dB);
  gsl_powers<<<64, 128, 0, stream>>>(dA, Cmat, Wst, P);
  gsl_v     <<<dim3(SEQL / 16, NBATCH), 128, 0, stream>>>(dB, u, V);
  gsl_scan  <<<NCHK, 128, 0, stream>>>(dA, V, Sg);
  gsl_chunks<<<1, 256, 0, stream>>>(P, Sg, SinT);
  gsl_final <<<dim3(NCHK * 4, 2, NBATCH), 128, 0, stream>>>(u, Cmat, Dvec, Sg, SinT, Wst,
                                                            (float*)d_out);
}